// Group_34265249088347
// MI455X (gfx1250) — compile-verified
//
#include <hip/hip_runtime.h>

// Problem constants from the reference
#define B_  16
#define N_  8192
#define G_  512     // NUM_GROUP
#define K_  32      // GROUP_SIZE
#define FPS_THREADS 256
#define FINF 3.402823466e38f

typedef __attribute__((ext_vector_type(2))) float v2f;
typedef __attribute__((ext_vector_type(8))) float v8f;
typedef unsigned long long u64;

// ======================= wave32 cross-lane helpers =======================
// All xor-distance shuffles done in the VALU (DPP8 / permlane16 / permlanex16),
// no LDS pipe, no dscnt waits. EXEC is all-ones at every use site.

// xor within 8-lane group via DPP8 (sel[i] = i^M)
template <unsigned M>
__device__ __forceinline__ unsigned dpp8_xor(unsigned v) {
  constexpr int imm = (int)(((0u ^ M) << 0)  | ((1u ^ M) << 3)  | ((2u ^ M) << 6)  |
                            ((3u ^ M) << 9)  | ((4u ^ M) << 12) | ((5u ^ M) << 15) |
                            ((6u ^ M) << 18) | ((7u ^ M) << 21));
  return (unsigned)__builtin_amdgcn_mov_dpp8((int)v, imm);
}
// xor 8 within 16-lane row: permlane16 with sel[i] = i^8
__device__ __forceinline__ unsigned pl16_xor8(unsigned v) {
  return (unsigned)__builtin_amdgcn_permlane16((int)v, (int)v,
                                               0xFEDCBA98u, 0x76543210u,
                                               false, false);
}
// xor 16: permlanex16, identity selects -> lane i reads lane i^16
__device__ __forceinline__ unsigned plx16_xor16(unsigned v) {
  return (unsigned)__builtin_amdgcn_permlanex16((int)v, (int)v,
                                                0x76543210u, 0xFEDCBA98u,
                                                false, false);
}
// xor 31 (full reversal): permlanex16, sel[i] = 15-i -> lane i reads lane 31-i
__device__ __forceinline__ unsigned plx16_rev(unsigned v) {
  return (unsigned)__builtin_amdgcn_permlanex16((int)v, (int)v,
                                                0x89ABCDEFu, 0x01234567u,
                                                false, false);
}
template <int M>
__device__ __forceinline__ unsigned xorshuf32(unsigned v) {
  if constexpr (M == 1 || M == 2 || M == 4) return dpp8_xor<(unsigned)M>(v);
  else if constexpr (M == 8)  return pl16_xor8(v);
  else if constexpr (M == 16) return plx16_xor16(v);
  else                        return plx16_rev(v);   // M == 31
}
template <int M>
__device__ __forceinline__ u64 swzx64(u64 v) {
  unsigned lo = xorshuf32<M>((unsigned)v);
  unsigned hi = xorshuf32<M>((unsigned)(v >> 32));
  return ((u64)hi << 32) | lo;
}
__device__ __forceinline__ u64 readlane64_31(u64 v) {
  unsigned lo = (unsigned)__builtin_amdgcn_readlane((int)(unsigned)v, 31);
  unsigned hi = (unsigned)__builtin_amdgcn_readlane((int)(unsigned)(v >> 32), 31);
  return ((u64)hi << 32) | lo;
}
// monotonic uint mapping: preserves float ordering for all finite values
__device__ __forceinline__ unsigned mono_f32(float f) {
  unsigned u = __float_as_uint(f);
  return u ^ ((unsigned)((int)u >> 31) | 0x80000000u);
}

// =====================================================================
// Kernel 1: FPS. One workgroup per batch (serial in G by construction).
// Argmax carried as packed u64 key: (raw dist bits << 32) | (N-1-i) so a
// branchless u64 max picks largest dist, lowest index on ties (jnp.argmax).
// =====================================================================
__global__ __launch_bounds__(FPS_THREADS) void fps_kernel(
    const float* __restrict__ xyz,     // [B,N,3]
    int*   __restrict__ cidx_out,      // [B,G]
    float* __restrict__ center_out) {  // [B,G,3]
  const int b   = blockIdx.x;
  const int tid = threadIdx.x;

  __shared__ float s_d[N_];            // running min sq-dist, 32KB
  __shared__ u64   s_w[FPS_THREADS / 32];
  __shared__ int   s_far;

  for (int i = tid; i < N_; i += FPS_THREADS) s_d[i] = FINF;
  if (tid == 0) s_far = 0;
  __syncthreads();

  const float* xb = xyz + (size_t)b * N_ * 3;

  #pragma unroll 1
  for (int it = 0; it < G_; ++it) {
    const int far = s_far;
    const float cx = xb[far * 3 + 0];
    const float cy = xb[far * 3 + 1];
    const float cz = xb[far * 3 + 2];
    if (tid == 0) {
      cidx_out[b * G_ + it] = far;
      center_out[((size_t)b * G_ + it) * 3 + 0] = cx;
      center_out[((size_t)b * G_ + it) * 3 + 1] = cy;
      center_out[((size_t)b * G_ + it) * 3 + 2] = cz;
    }

    u64 best = 0;                       // dist >= 0 -> raw bits are monotonic
    #pragma unroll 1
    for (int i = tid; i < N_; i += FPS_THREADS) {
      const float dx = xb[i * 3 + 0] - cx;
      const float dy = xb[i * 3 + 1] - cy;
      const float dz = xb[i * 3 + 2] - cz;
      float d = dx * dx + dy * dy + dz * dz;
      d = fminf(s_d[i], d);
      s_d[i] = d;
      const u64 key = ((u64)__float_as_uint(d) << 32) | (unsigned)(N_ - 1 - i);
      best = (key > best) ? key : best;
    }
    // wave32 max-reduce, all in the VALU
    { const u64 o = swzx64<16>(best); best = (o > best) ? o : best; }
    { const u64 o = swzx64< 8>(best); best = (o > best) ? o : best; }
    { const u64 o = swzx64< 4>(best); best = (o > best) ? o : best; }
    { const u64 o = swzx64< 2>(best); best = (o > best) ? o : best; }
    { const u64 o = swzx64< 1>(best); best = (o > best) ? o : best; }
    if ((tid & 31) == 0) s_w[tid >> 5] = best;
    __syncthreads();
    if (tid == 0) {
      u64 fb = s_w[0];
      #pragma unroll
      for (int w = 1; w < FPS_THREADS / 32; ++w) {
        const u64 o = s_w[w];
        fb = (o > fb) ? o : fb;
      }
      s_far = N_ - 1 - (int)(unsigned)fb;
    }
    __syncthreads();
  }
}

// ---- branchless bitonic stages on u64 keys (keys unique -> strict cmp safe) ----
#define SORT_STAGE(K, J)                                                     \
  {                                                                          \
    const u64  o  = swzx64<(J)>(key);                                        \
    const bool tm = ((lane & (K)) == 0) == ((lane & (J)) == 0);              \
    const bool sw = (o < key) == tm;                                         \
    key = sw ? o : key;                                                      \
  }
#define CLEAN_STAGE(J)                                                       \
  {                                                                          \
    const u64  o  = swzx64<(J)>(lv);                                         \
    const bool sw = (o < lv) == ((lane & (J)) == 0);                         \
    lv = sw ? o : lv;                                                        \
  }

// =====================================================================
// Kernel 2: kNN. One wave owns 16 query rows. Per loop iteration TWO
// V_WMMA_F32_16X16X4_F32 produce a 16x32 tile of ordering keys
// D[m][n] = ||r_n||^2 - 2 q_m . r_n  (row-constant ||q||^2 dropped).
// Top-32 per row held as sorted u64 keys (mono(dist)<<13 | idx), one per
// lane; merged via branchless VALU bitonic networks with a
// scalar-threshold early-out per 32 columns.
// =====================================================================
__global__ __launch_bounds__(32) void knn_kernel(
    const float* __restrict__ xyz,     // [B,N,3]
    const float* __restrict__ center,  // [B,G,3]
    int*  __restrict__ idx_out) {      // [B,G,K]
  const int wg   = blockIdx.x;         // 0 .. B*(G/16)-1
  const int b    = wg >> 5;            // / (G/16)
  const int g0   = (wg & 31) * 16;
  const int lane = threadIdx.x;        // wave32
  const int m    = lane & 15;
  const bool hi  = lane >= 16;

  __shared__ float tile[16 * 32];      // 16 rows x 32 cols, 2KB

  const float* xb = xyz + (size_t)b * N_ * 3;
  const float* qb = center + ((size_t)b * G_ + g0) * 3;

  // A 16x4 f32: lane<16 holds (K0,K1), lane>=16 holds (K2,K3); M = lane&15
  // row m = (-2qx, -2qy, -2qz, 1)
  const float qx = qb[m * 3 + 0];
  const float qy = qb[m * 3 + 1];
  const float qz = qb[m * 3 + 2];
  v2f A;
  A.x = hi ? (-2.0f * qz) : (-2.0f * qx);
  A.y = hi ? 1.0f         : (-2.0f * qy);

  // per-row sorted top-32: element i of row r lives in lane i
  u64 list[16];
  #pragma unroll
  for (int r = 0; r < 16; ++r) list[r] = ~0ull;

  #pragma unroll 1
  for (int n0 = 0; n0 < N_; n0 += 32) {
    // B column n = (rx, ry, rz, ||r||^2); two 16-col tiles
    #pragma unroll
    for (int t = 0; t < 2; ++t) {
      const int   n  = n0 + t * 16 + m;
      const float rx = xb[n * 3 + 0];
      const float ry = xb[n * 3 + 1];
      const float rz = xb[n * 3 + 2];
      const float rr = rx * rx + ry * ry + rz * rz;
      v2f Bm;
      Bm.x = hi ? rz : rx;
      Bm.y = hi ? rr : ry;
      v8f C = {0.f, 0.f, 0.f, 0.f, 0.f, 0.f, 0.f, 0.f};
      v8f D = __builtin_amdgcn_wmma_f32_16x16x4_f32(
          false, A, false, Bm, (short)0, C, false, false);
      // D layout: VGPR j -> row j (lanes 0-15) / row j+8 (lanes 16-31); col = lane&15
      if (t == 0) __syncthreads();     // prior tile reads done before overwrite
      #pragma unroll
      for (int j = 0; j < 8; ++j)
        tile[(j + (hi ? 8 : 0)) * 32 + t * 16 + m] = D[j];
    }
    __syncthreads();

    #pragma unroll
    for (int r = 0; r < 16; ++r) {
      // candidate key: unique (idx in low 13 bits) -> strict compares safe,
      // ties resolve to lowest index exactly like lax.top_k
      u64 key = ((u64)mono_f32(tile[r * 32 + lane]) << 13) | (unsigned)(n0 + lane);
      const u64 thresh = readlane64_31(list[r]);        // current 32nd best
      if (!__any(key < thresh)) continue;               // uniform skip

      // bitonic sort of 32 candidate keys, ascending (15 stages)
      SORT_STAGE( 2,  1)
      SORT_STAGE( 4,  2) SORT_STAGE( 4,  1)
      SORT_STAGE( 8,  4) SORT_STAGE( 8,  2) SORT_STAGE( 8,  1)
      SORT_STAGE(16,  8) SORT_STAGE(16,  4) SORT_STAGE(16,  2) SORT_STAGE(16, 1)
      SORT_STAGE(32, 16) SORT_STAGE(32,  8) SORT_STAGE(32,  4) SORT_STAGE(32, 2)
      SORT_STAGE(32,  1)

      // merge: reversed candidates vs sorted list -> keep 32 smallest
      const u64 rv = swzx64<31>(key);                   // lane 31-i, one permlanex16
      u64 lv = list[r];
      lv = (rv < lv) ? rv : lv;
      // bitonic clean, ascending (5 stages)
      CLEAN_STAGE(16) CLEAN_STAGE(8) CLEAN_STAGE(4) CLEAN_STAGE(2) CLEAN_STAGE(1)
      list[r] = lv;
    }
  }

  // lane i holds i-th nearest neighbor of each row; idx = low 13 bits
  #pragma unroll
  for (int r = 0; r < 16; ++r)
    idx_out[((size_t)b * G_ + g0 + r) * K_ + lane] = (int)(list[r] & (u64)(N_ - 1));
}

// =====================================================================
// Outputs concatenated flat in reference return order:
//   idx [B,G,K] int32 | center_idx [B,G] int32 | center [B,G,3] f32
// =====================================================================
extern "C" void kernel_launch(void* const* d_in, const int* in_sizes, int n_in,
                              void* d_out, int out_size, void* d_ws, size_t ws_size,
                              hipStream_t stream) {
  (void)in_sizes; (void)n_in; (void)out_size; (void)d_ws; (void)ws_size;
  const float* xyz = (const float*)d_in[0];

  int*   idx_out    = (int*)d_out;                              // B*G*K ints
  int*   cidx_out   = idx_out + (size_t)B_ * G_ * K_;           // B*G ints
  float* center_out = (float*)(cidx_out + (size_t)B_ * G_);     // B*G*3 floats

  fps_kernel<<<B_, FPS_THREADS, 0, stream>>>(xyz, cidx_out, center_out);
  knn_kernel<<<B_ * (G_ / 16), 32, 0, stream>>>(xyz, center_out, idx_out);
}